// ResidualVectorQuantizer_54004918780623
// MI455X (gfx1250) — compile-verified
//
#include <hip/hip_runtime.h>
#include <hip/hip_bf16.h>
#include <math.h>

typedef float v2f __attribute__((ext_vector_type(2)));
typedef float v8f __attribute__((ext_vector_type(8)));
typedef unsigned int uint32x4 __attribute__((ext_vector_type(4)));
typedef int int32x4 __attribute__((ext_vector_type(4)));
typedef int int32x8 __attribute__((ext_vector_type(8)));

#define NCB 4
#define CBK 256
#define DIM 64
#define BT_TOK (64 * 4096)
#define NBLK (BT_TOK / 16)
#define NWAVES 4
#define TPB (NWAVES * 32)
#define GRID 256

__device__ __forceinline__ v8f wmma_f32_k4(v2f a, v2f b, v8f c) {
  // D = A(16x4, f32) * B(4x16, f32) + C(16x16, f32)
  return __builtin_amdgcn_wmma_f32_16x16x4_f32(false, a, false, b, (short)0, c,
                                               false, false);
}

__launch_bounds__(TPB)
__global__ void rvq_fused_kernel(const float* __restrict__ z,
                                 const float* __restrict__ w2v,
                                 const float* __restrict__ Eg,
                                 const float* __restrict__ W1,
                                 const float* __restrict__ b1,
                                 const float* __restrict__ W2,
                                 const float* __restrict__ b2,
                                 float* __restrict__ zq_out,
                                 int* __restrict__ codes_out,
                                 int* __restrict__ codes0_out,
                                 float* __restrict__ partials) {
  __shared__ float sE[NCB * CBK * DIM];   // 256 KB: all 4 codebooks, TDM-staged
  __shared__ float sScale[NCB * CBK];     // 0.5*||e||^2 per code (4 KB)
  __shared__ float sW1t[DIM * DIM];       // W1^T: [out][in] (16 KB)
  __shared__ float sW2t[DIM * DIM];       // W2^T: [out][in] (16 KB)
  __shared__ float sB1[DIM];
  __shared__ float sB2[DIM];
  __shared__ int sCodes[NWAVES][16];      // per-wave code redistribute
  __shared__ float sH[NWAVES][16 * DIM];  // per-wave h transpose scratch (16 KB)
  __shared__ float sRed[TPB];

  const int tid = threadIdx.x;

  // ---- Tensor Data Mover: DMA all 4 codebooks (256 KB) into LDS with a
  // single 1D descriptor: 32768 elements of 8 bytes. Wave 0 issues it.
  if (tid < 32) {
    const unsigned long long ga = (unsigned long long)Eg;
    const unsigned lds_addr = (unsigned)(unsigned long long)(&sE[0]);
    uint32x4 g0;
    g0.x = 0x1u;                            // count=1, user-mode descriptor
    g0.y = lds_addr;                        // LDS byte address
    g0.z = (unsigned)(ga & 0xFFFFFFFFu);    // global_addr[31:0]
    g0.w = (unsigned)((ga >> 32) & 0x1FFFFFFu) | (2u << 30);  // addr hi, type=2
    int32x8 g1;
    g1[0] = 0x30000;                  // workgroup_mask=0, data_size=3 (8B)
    g1[1] = (int)(0x8000u << 16);     // tensor_dim0 = 32768 (low16 @ [31:16])
    g1[2] = 0;                        // tensor_dim0 hi, tensor_dim1 lo
    g1[3] = (int)(0x8000u << 16);     // tile_dim0 = 32768 @ [31:16]
    g1[4] = 0;                        // tile_dim1=0 (1D), tile_dim2=0
    g1[5] = 0x8000;                   // tensor_dim0_stride low (unused for 1D)
    g1[6] = 0;
    g1[7] = 0;
    int32x4 gz = {0, 0, 0, 0};
#if __clang_major__ >= 23
    int32x8 gz8 = {0, 0, 0, 0, 0, 0, 0, 0};
    __builtin_amdgcn_tensor_load_to_lds(g0, g1, gz, gz, gz8, 0);
#else
    __builtin_amdgcn_tensor_load_to_lds(g0, g1, gz, gz, 0);
#endif
    __builtin_amdgcn_s_wait_tensorcnt(0);
  }

  // Stage W1/W2 transposed ([out][in]) so B-operand k,k+1 pairs are contiguous.
  for (int i = tid; i < DIM * DIM; i += TPB) {
    const int r = i >> 6;    // input dim
    const int c = i & 63;    // output dim
    sW1t[c * DIM + r] = W1[r * DIM + c];
    sW2t[c * DIM + r] = W2[r * DIM + c];
  }
  if (tid < DIM) {
    sB1[tid] = b1[tid];
    sB2[tid] = b2[tid];
  }
  __syncthreads();  // E (TDM), W, b all resident in LDS

  // Precompute 0.5*||e||^2 for all 1024 codes from LDS-resident E.
  for (int c = tid; c < NCB * CBK; c += TPB) {
    const float* e = sE + c * DIM;
    float s = 0.f;
    for (int d = 0; d < DIM; d += 2) {
      v2f v = *(const v2f*)(e + d);
      s += v.x * v.x + v.y * v.y;
    }
    sScale[c] = 0.5f * s;
  }
  __syncthreads();

  const int lane = tid & 31;
  const int wave = tid >> 5;
  const int half = lane >> 4;   // 0: K pair {0,1}, 1: K pair {2,3} (A/B layout)
  const int lrow = lane & 15;   // matrix row / column within tile
  const int kofs = half * 2;

  float vqAcc = 0.f, semAcc = 0.f;

  for (int blk = blockIdx.x * NWAVES + wave; blk < NBLK;
       blk += GRID * NWAVES) {
    const int rowBase = blk * 16;
    __builtin_prefetch(z + (rowBase + GRID * NWAVES * 16) * DIM, 0, 0);

    // Residual in WMMA A-matrix layout: ra[kk] covers K = kk*4+kofs..+1
    v2f ra[16];
#pragma unroll
    for (int kk = 0; kk < 16; ++kk)
      ra[kk] = *(const v2f*)(z + (rowBase + lrow) * DIM + kk * 4 + kofs);

    for (int cb = 0; cb < NCB; ++cb) {
      const float* __restrict__ Ecb = sE + cb * CBK * DIM;
      const float* __restrict__ Scb = sScale + cb * CBK;

      float bestV[8];
      int bestI[8];
#pragma unroll
      for (int r = 0; r < 8; ++r) {
        bestV[r] = -3.4e38f;
        bestI[r] = 0;
      }

      // Scores = residual @ E^T, tile by tile (16 codes per tile), B from LDS
      for (int j = 0; j < 16; ++j) {
        v8f acc = {0.f, 0.f, 0.f, 0.f, 0.f, 0.f, 0.f, 0.f};
        const int code = j * 16 + lrow;  // this lane's column (B layout)
        const float* __restrict__ erow = Ecb + code * DIM + kofs;
#pragma unroll
        for (int kk = 0; kk < 16; ++kk) {
          v2f bt = *(const v2f*)(erow + kk * 4);
          acc = wmma_f32_k4(ra[kk], bt, acc);
        }
        const float sc = Scb[code];
#pragma unroll
        for (int r = 0; r < 8; ++r) {
          const float s = acc[r] - sc;  // argmax s <=> argmin L2 dist
          if (s > bestV[r] || (s == bestV[r] && code < bestI[r])) {
            bestV[r] = s;
            bestI[r] = code;
          }
        }
      }

      // Argmax reduction across each 16-lane half (rows r / r+8)
#pragma unroll
      for (int r = 0; r < 8; ++r) {
#pragma unroll
        for (int m = 1; m < 16; m <<= 1) {
          const float ov = __shfl_xor(bestV[r], m, 32);
          const int oi = __shfl_xor(bestI[r], m, 32);
          if (ov > bestV[r] || (ov == bestV[r] && oi < bestI[r])) {
            bestV[r] = ov;
            bestI[r] = oi;
          }
        }
      }
#pragma unroll
      for (int r = 0; r < 8; ++r) {
        if (lrow == r) {
          const int row = r + 8 * half;
          const int c = bestI[r];
          sCodes[wave][row] = c;
          codes_out[(rowBase + row) * NCB + cb] = c;
          if (cb == 0) codes0_out[rowBase + row] = c;
        }
      }
      __builtin_amdgcn_wave_barrier();
      const int myCode = sCodes[wave][lrow];
      const float* __restrict__ qrow = Ecb + myCode * DIM + kofs;

      if (cb == 0) {
        // Fused: gather z_q0, update residual + vq loss, and run
        // h = gelu(z_q0 @ W1 + b1) with z_q0 as the WMMA A operand.
        v8f hacc[4];
#pragma unroll
        for (int j = 0; j < 4; ++j) {
          const float bv = sB1[j * 16 + lrow];
#pragma unroll
          for (int r = 0; r < 8; ++r) hacc[j][r] = bv;
        }
#pragma unroll
        for (int kk = 0; kk < 16; ++kk) {
          v2f q = *(const v2f*)(qrow + kk * 4);
          v2f dd = ra[kk] - q;
          vqAcc += dd.x * dd.x + dd.y * dd.y;
          ra[kk] = ra[kk] - q;
#pragma unroll
          for (int j = 0; j < 4; ++j) {
            v2f wt = *(const v2f*)(&sW1t[(j * 16 + lrow) * DIM + kk * 4 + kofs]);
            hacc[j] = wmma_f32_k4(q, wt, hacc[j]);
          }
        }
        // exact gelu + D-layout -> A-layout transpose via LDS
#pragma unroll
        for (int j = 0; j < 4; ++j)
#pragma unroll
          for (int r = 0; r < 8; ++r) {
            const float x = hacc[j][r];
            sH[wave][(r + 8 * half) * DIM + j * 16 + lrow] =
                0.5f * x * (1.f + erff(x * 0.70710678118654752f));
          }
        __builtin_amdgcn_wave_barrier();
        // pred = h @ W2 + b2 ; semantic MSE vs w2v targets
#pragma unroll
        for (int j = 0; j < 4; ++j) {
          const float bv = sB2[j * 16 + lrow];
          v8f p;
#pragma unroll
          for (int r = 0; r < 8; ++r) p[r] = bv;
#pragma unroll
          for (int kk = 0; kk < 16; ++kk) {
            v2f ha = *(const v2f*)(&sH[wave][lrow * DIM + kk * 4 + kofs]);
            v2f wt = *(const v2f*)(&sW2t[(j * 16 + lrow) * DIM + kk * 4 + kofs]);
            p = wmma_f32_k4(ha, wt, p);
          }
#pragma unroll
          for (int r = 0; r < 8; ++r) {
            const float t =
                w2v[(rowBase + r + 8 * half) * DIM + j * 16 + lrow];
            const float d = p[r] - t;
            semAcc += d * d;
          }
        }
      } else {
        // gather z_q, residual update + vq loss only
#pragma unroll
        for (int kk = 0; kk < 16; ++kk) {
          v2f q = *(const v2f*)(qrow + kk * 4);
          v2f dd = ra[kk] - q;
          vqAcc += dd.x * dd.x + dd.y * dd.y;
          ra[kk] = ra[kk] - q;
        }
      }
    }

    // z_q_total = z - residual_final (forward identity)
#pragma unroll
    for (int kk = 0; kk < 16; ++kk) {
      v2f z0 = *(const v2f*)(z + (rowBase + lrow) * DIM + kk * 4 + kofs);
      v2f q = z0 - ra[kk];
      *(v2f*)(zq_out + (rowBase + lrow) * DIM + kk * 4 + kofs) = q;
    }
  }

  // Deterministic block reduction -> per-block partials
  sRed[tid] = vqAcc;
  __syncthreads();
  for (int s = TPB / 2; s > 0; s >>= 1) {
    if (tid < s) sRed[tid] += sRed[tid + s];
    __syncthreads();
  }
  if (tid == 0) partials[blockIdx.x * 2 + 0] = sRed[0];
  __syncthreads();
  sRed[tid] = semAcc;
  __syncthreads();
  for (int s = TPB / 2; s > 0; s >>= 1) {
    if (tid < s) sRed[tid] += sRed[tid + s];
    __syncthreads();
  }
  if (tid == 0) partials[blockIdx.x * 2 + 1] = sRed[0];
}

__global__ void rvq_reduce_kernel(const float* __restrict__ partials,
                                  float* __restrict__ vq_out,
                                  float* __restrict__ sem_out) {
  if (blockIdx.x == 0 && threadIdx.x == 0) {
    float v = 0.f, s = 0.f;
    for (int i = 0; i < GRID; ++i) {
      v += partials[2 * i];
      s += partials[2 * i + 1];
    }
    const float inv = 1.0f / ((float)BT_TOK * (float)DIM);
    *vq_out = 1.25f * v * inv;  // emb_loss + 0.25*commit (same fwd value)
    *sem_out = s * inv;
  }
}

extern "C" void kernel_launch(void* const* d_in, const int* in_sizes, int n_in,
                              void* d_out, int out_size, void* d_ws,
                              size_t ws_size, hipStream_t stream) {
  const float* z = (const float*)d_in[0];
  const float* w2v = (const float*)d_in[1];
  const float* E = (const float*)d_in[2];
  const float* W1 = (const float*)d_in[3];
  const float* b1 = (const float*)d_in[4];
  const float* W2 = (const float*)d_in[5];
  const float* b2 = (const float*)d_in[6];

  float* out = (float*)d_out;
  float* zq_out = out;                                          // [B,T,64]
  int* codes_out = (int*)(out + BT_TOK * DIM);                  // [B,T,4]
  int* codes0_out = (int*)(out + BT_TOK * DIM + BT_TOK * NCB);  // [B,T]
  float* vq_out = out + BT_TOK * DIM + BT_TOK * NCB + BT_TOK;
  float* sem_out = vq_out + 1;

  float* partials = (float*)d_ws;  // [GRID][2]

  rvq_fused_kernel<<<GRID, TPB, 0, stream>>>(z, w2v, E, W1, b1, W2, b2, zq_out,
                                             codes_out, codes0_out, partials);
  rvq_reduce_kernel<<<1, 32, 0, stream>>>(partials, vq_out, sem_out);
}